// InterpolativeUpsampler_12472585027918
// MI455X (gfx1250) — compile-verified
//
#include <hip/hip_runtime.h>

// Problem constants (from the reference: x [8,128,256,256] f32, kernel [4,4], up=2)
#define NPLANES (8 * 128)
#define IH 256
#define IW 256
#define OH 512
#define OW 512

// Tile: 16x16 threads, each thread -> 2x4 output block => 32x64 output tile
// => 16x32 input tile + 1 halo each side => 18x34 floats in LDS.
#define LDSH 18
#define LDSW 34
#define LDSP 36  // padded row stride (floats) to avoid bank conflicts

typedef float v4f __attribute__((ext_vector_type(4)));
typedef __attribute__((address_space(1))) int* as1_i32p;
typedef __attribute__((address_space(3))) int* as3_i32p;

#define HAVE_ASYNC_LDS __has_builtin(__builtin_amdgcn_global_load_async_to_lds_b32)

__global__ __launch_bounds__(256) void upsample2d_fir4_kernel(
    const float* __restrict__ x, const float* __restrict__ k4,
    float* __restrict__ out) {
  __shared__ float tile[LDSH * LDSP];
  __shared__ float Wk[16];

  const int t  = threadIdx.x;
  const int tx = t & 15;
  const int ty = t >> 4;
  const int bx = blockIdx.x;           // 0..7   (output tiles of 64 cols)
  const int by = blockIdx.y;           // 0..15  (output tiles of 32 rows)
  const int p  = blockIdx.z;           // plane = n*C + c, 0..1023

  const int inX0 = bx * 32 - 1;        // global input col of LDS col 0
  const int inY0 = by * 16 - 1;        // global input row of LDS row 0

  const float* __restrict__ xin = x + (size_t)p * (IH * IW);

  // ---- Phase 0: stage flipped, gain-premultiplied weights; zero halo on edge tiles
  if (t < 16) Wk[t] = 4.0f * k4[15 - t];  // W = up^2 * flip(k); symmetric anyway
  const bool edge = (bx == 0) | (bx == 7) | (by == 0) | (by == 15);
  if (edge) {
    for (int e = t; e < LDSH * LDSP; e += 256) tile[e] = 0.0f;
  }
  __syncthreads();

  // ---- Phase 1: async global -> LDS staging of the in-range input tile
  for (int e = t; e < LDSH * LDSW; e += 256) {
    const int r  = e / LDSW;
    const int c  = e - r * LDSW;
    const int gr = inY0 + r;
    const int gc = inX0 + c;
    if ((unsigned)gr < (unsigned)IH && (unsigned)gc < (unsigned)IW) {
      const float* src = xin + gr * IW + gc;
      float* dst = &tile[r * LDSP + c];
#if HAVE_ASYNC_LDS
      __builtin_amdgcn_global_load_async_to_lds_b32(
          (as1_i32p)src, (as3_i32p)dst, 0, 0);
#else
      *dst = *src;
#endif
    }
  }
#if HAVE_ASYNC_LDS
#if __has_builtin(__builtin_amdgcn_s_wait_asynccnt)
  __builtin_amdgcn_s_wait_asynccnt(0);
#else
  asm volatile("s_wait_asynccnt 0" ::: "memory");
#endif
#endif
  __syncthreads();

  // ---- Phase 2: per-thread 3x4 stencil from LDS -> 2x4 outputs
  float W[4][4];
#pragma unroll
  for (int i = 0; i < 16; ++i) W[i >> 2][i & 3] = Wk[i];

  float s[3][4];
#pragma unroll
  for (int a = 0; a < 3; ++a) {
    const float2 p0 = *(const float2*)&tile[(ty + a) * LDSP + 2 * tx];
    const float2 p1 = *(const float2*)&tile[(ty + a) * LDSP + 2 * tx + 2];
    s[a][0] = p0.x; s[a][1] = p0.y; s[a][2] = p1.x; s[a][3] = p1.y;
  }

  // out(2m+py, 2n'+px), n' = n+u :
  //   sum_{a,b} W[2a+py][2b+px] * s[a+py][u+b+px]
  float o[2][4];
#pragma unroll
  for (int py = 0; py < 2; ++py) {
#pragma unroll
    for (int u = 0; u < 2; ++u) {
#pragma unroll
      for (int px = 0; px < 2; ++px) {
        float acc = 0.0f;
#pragma unroll
        for (int a = 0; a < 2; ++a) {
#pragma unroll
          for (int b = 0; b < 2; ++b) {
            acc = fmaf(W[2 * a + py][2 * b + px], s[a + py][u + b + px], acc);
          }
        }
        o[py][2 * u + px] = acc;
      }
    }
  }

  // ---- Phase 3: two aligned 128-bit non-temporal stores (write-once data)
  const int oy0 = by * 32 + 2 * ty;
  const int ox0 = bx * 64 + 4 * tx;
  float* op = out + ((size_t)p * OH + oy0) * OW + ox0;

  v4f r0 = {o[0][0], o[0][1], o[0][2], o[0][3]};
  v4f r1 = {o[1][0], o[1][1], o[1][2], o[1][3]};
  __builtin_nontemporal_store(r0, (v4f*)op);
  __builtin_nontemporal_store(r1, (v4f*)(op + OW));
}

extern "C" void kernel_launch(void* const* d_in, const int* in_sizes, int n_in,
                              void* d_out, int out_size, void* d_ws, size_t ws_size,
                              hipStream_t stream) {
  const float* x  = (const float*)d_in[0];   // [8,128,256,256] f32
  const float* k4 = (const float*)d_in[1];   // [4,4] f32
  float* out = (float*)d_out;                // [8,128,512,512] f32

  dim3 grid(OW / 64, OH / 32, NPLANES);      // (8, 16, 1024)
  upsample2d_fir4_kernel<<<grid, 256, 0, stream>>>(x, k4, out);
}